// FrequencySelectiveFilter_71906342470090
// MI455X (gfx1250) — compile-verified
//
#include <hip/hip_runtime.h>

typedef __attribute__((ext_vector_type(2))) float v2f;
typedef __attribute__((ext_vector_type(8))) float v8f;

#define BATCH 16
#define SEQ   1024
#define DMOD  512
#define DN    32
#define CH    16
#define NCH   (SEQ / CH)

static constexpr size_t TOT = (size_t)BATCH * SEQ * DMOD;

// D = A(16x4 f32) * B(4x16 f32) + C, full-precision fp32 WMMA.
// 8-arg form: (neg_a, A, neg_b, B, c_mod, C, reuse_a, reuse_b)
static __device__ __forceinline__ v8f wmma4(v2f a, v2f b, v8f c) {
  return __builtin_amdgcn_wmma_f32_16x16x4_f32(false, a, false, b, (short)0, c,
                                               false, false);
}

// Half-wave swap: lane L <-> lane L^16. Pure VALU via v_permlanex16_b32
// (identity nibble selects make each lane read the opposite 16-lane row's
// same-position lane); falls back to shuffle if the builtin is missing.
static __device__ __forceinline__ float xor16(float v) {
#if __has_builtin(__builtin_amdgcn_permlanex16)
  int i = __builtin_bit_cast(int, v);
  int r = __builtin_amdgcn_permlanex16(i, i, 0x76543210, 0xfedcba98, false, false);
  return __builtin_bit_cast(float, r);
#else
  return __shfl_xor(v, 16, 32);
#endif
}

static __device__ __forceinline__ v8f zero8() {
  v8f z;
#pragma unroll
  for (int i = 0; i < 8; ++i) z[i] = 0.0f;
  return z;
}

// One wave32 per (channel d, ssm). Chunked scan: per 16-step chunk,
//   Y^T[t,b]  = Ktri(t,s) @ X^T(s,b) + G(t,n) @ H^T(n,b)
//   H^T(n,b) <- dA^16 (.) H^T + Q^T(n,s) @ X^T(s,b)
// Ktri, G, Q^T, dA^16 are chunk-invariant (time-invariant dA) -> precomputed.
__global__ __launch_bounds__(32)
void ssm_chunkscan_wmma(const float* __restrict__ x,
                        const float* __restrict__ Alog0, const float* __restrict__ B0,
                        const float* __restrict__ C0, const float* __restrict__ ldt0,
                        const float* __restrict__ w0,
                        const float* __restrict__ Alog1, const float* __restrict__ B1,
                        const float* __restrict__ C1, const float* __restrict__ ldt1,
                        const float* __restrict__ w1,
                        float* __restrict__ out) {
  const int d = blockIdx.x;
  const int ssm = blockIdx.y;
  const int ln = threadIdx.x;
  const bool lolane = ln < 16;
  const int m16 = ln & 15;       // A-op row (t or n) / B,D-op column (b)
  const int so = lolane ? 0 : 2; // K-slot offset within a K=4 block
  const int nb = lolane ? 0 : 8; // D-layout row offset

  const float* Alog = ssm ? Alog1 : Alog0;
  const float* Bp = ssm ? B1 : B0;
  const float* Cp = ssm ? C1 : C0;
  const float* ldtp = ssm ? ldt1 : ldt0;
  const float* wp = ssm ? w1 : w0;

  __shared__ float shLA[DN], shC[DN], shB[DN];

  // lane == state index n for parameter setup
  const float dt = expf(ldtp[d]);
  const float lA = dt * expf(Alog[(size_t)d * DN + ln]); // log(dA)
  const float dA = expf(lA);
  const float dB = dt * Bp[(size_t)d * DN + ln];
  const float Cc = Cp[(size_t)d * DN + ln] * wp[d]; // fold weight into C

  shLA[ln] = lA;
  shB[ln] = dB;
  shC[ln] = Cc;
  __syncthreads();

  // ---- chunk-invariant operands ----
  // Ktri A-operand (M=t, K=s): K[t-s] for t>=s else 0, K[tau]=sum_n C dB dA^tau
  v2f ktriA[4];
#pragma unroll
  for (int j = 0; j < 4; ++j) { ktriA[j].x = 0.0f; ktriA[j].y = 0.0f; }
  {
    float p = 1.0f; // dA^tau
    for (int tau = 0; tau < CH; ++tau) {
      float v = Cc * dB * p;
#pragma unroll
      for (int off = 16; off > 0; off >>= 1) v += __shfl_xor(v, off, 32);
#pragma unroll
      for (int j = 0; j < 4; ++j) {
        const int sx = 4 * j + so;
        if (m16 - sx == tau) ktriA[j].x = v;
        if (m16 - sx - 1 == tau) ktriA[j].y = v;
      }
      p *= dA;
    }
  }

  // G A-operand (M=t, K=n): C_n * dA_n^(t+1)
  v2f gA[8];
#pragma unroll
  for (int j = 0; j < 8; ++j) {
    const int n0 = 4 * j + so;
    gA[j].x = shC[n0] * expf(shLA[n0] * (float)(m16 + 1));
    gA[j].y = shC[n0 + 1] * expf(shLA[n0 + 1] * (float)(m16 + 1));
  }

  // Q^T A-operand (M=n, K=s): dB_n * dA_n^(15-s), two M-halves (n<16, n>=16)
  v2f qA0[4], qA1[4];
  {
    const float la0 = shLA[m16], db0 = shB[m16];
    const float la1 = shLA[m16 + 16], db1 = shB[m16 + 16];
#pragma unroll
    for (int j = 0; j < 4; ++j) {
      const int s0 = 4 * j + so;
      qA0[j].x = db0 * expf(la0 * (float)(15 - s0));
      qA0[j].y = db0 * expf(la0 * (float)(14 - s0));
      qA1[j].x = db1 * expf(la1 * (float)(15 - s0));
      qA1[j].y = db1 * expf(la1 * (float)(14 - s0));
    }
  }

  // chunk decay dA^16 in the D-layout of H^T
  float pd1[8], pd2[8];
#pragma unroll
  for (int r = 0; r < 8; ++r) {
    pd1[r] = expf(shLA[r + nb] * 16.0f);
    pd2[r] = expf(shLA[r + nb + 16] * 16.0f);
  }

  // H^T state kept in WMMA D-layout: hd1 rows n = nb..nb+7, hd2 rows 16+nb..
  float hd1[8], hd2[8];
#pragma unroll
  for (int r = 0; r < 8; ++r) { hd1[r] = 0.0f; hd2[r] = 0.0f; }

  const float* xb = x + (size_t)m16 * SEQ * DMOD + d;
  float* yb = out + (size_t)ssm * TOT + (size_t)m16 * SEQ * DMOD + d;

  // software-pipelined X^T B-operand loads (lane holds b=m16, its K-slots of s)
  v2f xB[4], xN[4];
#pragma unroll
  for (int j = 0; j < 4; ++j) {
    xB[j].x = xb[(size_t)(4 * j + so) * DMOD];
    xB[j].y = xb[(size_t)(4 * j + so + 1) * DMOD];
  }

  for (int c = 0; c < NCH; ++c) {
    // prefetch next chunk (clamped; redundant final load keeps EXEC uniform)
    {
      const int cn = (c + 1 < NCH) ? (c + 1) : c;
      const float* xc = xb + (size_t)cn * CH * DMOD;
#pragma unroll
      for (int j = 0; j < 4; ++j) {
        xN[j].x = xc[(size_t)(4 * j + so) * DMOD];
        xN[j].y = xc[(size_t)(4 * j + so + 1) * DMOD];
      }
    }

    // intra-chunk causal Toeplitz matmul
    v8f acc = zero8();
#pragma unroll
    for (int j = 0; j < 4; ++j) acc = wmma4(ktriA[j], xB[j], acc);

    // carried-state contribution: convert H^T D-layout -> B-operand with a
    // half-wave exchange (lanes L and L+16 hold the same b column)
    {
      float px, py;
      v2f hb;
      px = xor16(hd1[2]); py = xor16(hd1[3]);
      hb.x = lolane ? hd1[0] : px; hb.y = lolane ? hd1[1] : py;
      acc = wmma4(gA[0], hb, acc);
      px = xor16(hd1[6]); py = xor16(hd1[7]);
      hb.x = lolane ? hd1[4] : px; hb.y = lolane ? hd1[5] : py;
      acc = wmma4(gA[1], hb, acc);
      px = xor16(hd1[0]); py = xor16(hd1[1]);
      hb.x = lolane ? px : hd1[2]; hb.y = lolane ? py : hd1[3];
      acc = wmma4(gA[2], hb, acc);
      px = xor16(hd1[4]); py = xor16(hd1[5]);
      hb.x = lolane ? px : hd1[6]; hb.y = lolane ? py : hd1[7];
      acc = wmma4(gA[3], hb, acc);
      px = xor16(hd2[2]); py = xor16(hd2[3]);
      hb.x = lolane ? hd2[0] : px; hb.y = lolane ? hd2[1] : py;
      acc = wmma4(gA[4], hb, acc);
      px = xor16(hd2[6]); py = xor16(hd2[7]);
      hb.x = lolane ? hd2[4] : px; hb.y = lolane ? hd2[5] : py;
      acc = wmma4(gA[5], hb, acc);
      px = xor16(hd2[0]); py = xor16(hd2[1]);
      hb.x = lolane ? px : hd2[2]; hb.y = lolane ? py : hd2[3];
      acc = wmma4(gA[6], hb, acc);
      px = xor16(hd2[4]); py = xor16(hd2[5]);
      hb.x = lolane ? px : hd2[6]; hb.y = lolane ? py : hd2[7];
      acc = wmma4(gA[7], hb, acc);
    }

    // input -> next-state contribution (two n-halves), then diagonal decay
    v8f c0 = zero8(), c1 = zero8();
#pragma unroll
    for (int j = 0; j < 4; ++j) c0 = wmma4(qA0[j], xB[j], c0);
#pragma unroll
    for (int j = 0; j < 4; ++j) c1 = wmma4(qA1[j], xB[j], c1);
#pragma unroll
    for (int r = 0; r < 8; ++r) {
      hd1[r] = fmaf(pd1[r], hd1[r], c0[r]);
      hd2[r] = fmaf(pd2[r], hd2[r], c1[r]);
    }

    // store Y^T D-layout: acc[r] -> (t = r + nb, b = m16)
    float* yc = yb + (size_t)c * CH * DMOD;
#pragma unroll
    for (int r = 0; r < 8; ++r) yc[(size_t)(r + nb) * DMOD] = acc[r];

#pragma unroll
    for (int j = 0; j < 4; ++j) xB[j] = xN[j];
  }
}

__global__ void combine_add(const float* __restrict__ a, const float* __restrict__ b,
                            float* __restrict__ o, size_t n4) {
  size_t i = (size_t)blockIdx.x * blockDim.x + threadIdx.x;
  const float4* a4 = (const float4*)a;
  const float4* b4 = (const float4*)b;
  float4* o4 = (float4*)o;
  for (; i < n4; i += (size_t)gridDim.x * blockDim.x) {
    float4 va = a4[i], vb = b4[i];
    o4[i] = make_float4(va.x + vb.x, va.y + vb.y, va.z + vb.z, va.w + vb.w);
  }
}

extern "C" void kernel_launch(void* const* d_in, const int* in_sizes, int n_in,
                              void* d_out, int out_size, void* d_ws, size_t ws_size,
                              hipStream_t stream) {
  const float* x = (const float*)d_in[0];
  const float* Ah = (const float*)d_in[1];
  const float* Bh = (const float*)d_in[2];
  const float* Ch = (const float*)d_in[3];
  const float* dth = (const float*)d_in[4];
  const float* Ab = (const float*)d_in[5];
  const float* Bb = (const float*)d_in[6];
  const float* Cb = (const float*)d_in[7];
  const float* dtb = (const float*)d_in[8];
  const float* wh = (const float*)d_in[9];
  const float* wb = (const float*)d_in[10];
  float* out = (float*)d_out;

  dim3 grid(DMOD, 2);
  ssm_chunkscan_wmma<<<grid, 32, 0, stream>>>(x, Ah, Bh, Ch, dth, wh,
                                              Ab, Bb, Cb, dtb, wb, out);

  const size_t n4 = TOT / 4;
  const int blocks = (int)((n4 + 255) / 256);
  combine_add<<<blocks, 256, 0, stream>>>(out, out + TOT, out + 2 * TOT, n4);
}